// ResnetEncoderMatching_23545010716858
// MI455X (gfx1250) — compile-verified
//
#include <hip/hip_runtime.h>
#include <cstdint>

// Problem constants (from reference)
#define B_ 4
#define F_ 2
#define C_ 16
#define H_ 48
#define W_ 160
#define D_ 96
#define DCHUNK 8
static constexpr int HW_ = H_ * W_;
static constexpr float EPS_ = 1e-7f;

typedef uint32_t u32;
typedef __attribute__((ext_vector_type(4))) unsigned int u32x4;
typedef __attribute__((ext_vector_type(4))) int i32x4;
typedef __attribute__((ext_vector_type(8))) int i32x8;

// ---------------------------------------------------------------------------
// Kernel 1: build the count-averaged (pre-fill) cost volume  cost_pre[B,D,H,W]
//   grid = (D/DCHUNK, H, B), block = W (=160 threads = 5 wave32)
//   Stages the current-feature row (C_ x W_ = 10KB) into LDS via the
//   Tensor Data Mover (2D tile descriptor), then loops DCHUNK depth bins.
// ---------------------------------------------------------------------------
__global__ void __launch_bounds__(W_)
cost_volume_kernel(const float* __restrict__ cur,     // [B,C,H,W]
                   const float* __restrict__ lf,      // [B,F,C,H,W]
                   const float* __restrict__ poses,   // [B,F,4,4]
                   const float* __restrict__ Kmat,    // [B,4,4]
                   const float* __restrict__ invK,    // [B,4,4]
                   const float* __restrict__ bins,    // [D]
                   float* __restrict__ out_cost)      // [B,D,H,W] (cost region)
{
    __shared__ float s_cur[C_ * W_];

    const int w  = threadIdx.x;
    const int h  = blockIdx.y;
    const int b  = blockIdx.z;
    const int d0 = blockIdx.x * DCHUNK;

    // &cur[b][0][h][0]; channel rows are H*W apart
    const float* curRow = cur + ((size_t)b * C_ * H_ + h) * W_;

#if __has_builtin(__builtin_amdgcn_tensor_load_to_lds) && __has_builtin(__builtin_amdgcn_s_wait_tensorcnt)
    // --- TDM: DMA a 2D tile (tile_dim0 = W_ elems, tile_dim1 = C_ rows,
    //          row stride = H_*W_ elems, 4-byte data) into LDS. ---
    if (threadIdx.x == 0) {
        const uint64_t ga = (uint64_t)(uintptr_t)curRow;
        const u32 laddr = (u32)(uintptr_t)(void*)&s_cur[0];
        u32x4 g0;
        g0[0] = 1u;                                   // count=1, user descriptor
        g0[1] = laddr;                                // lds_addr (bytes)
        g0[2] = (u32)(ga & 0xFFFFFFFFu);              // global_addr[31:0]
        g0[3] = (u32)((ga >> 32) & 0x01FFFFFFu)       // global_addr[56:32]
              | (2u << 30);                           // type = 2 ("image")
        i32x8 g1;
        g1[0] = 0x00020000;                           // data_size=2 (4 bytes)
        g1[1] = (int)((u32)W_ << 16);                 // tensor_dim0 = 160
        g1[2] = (int)((u32)C_ << 16);                 // tensor_dim1 = 16
        g1[3] = (int)((u32)W_ << 16);                 // tile_dim0   = 160
        g1[4] = C_;                                   // tile_dim1   = 16
        g1[5] = HW_;                                  // tensor_dim0_stride = H*W
        g1[6] = 0;
        g1[7] = 0;
        i32x4 gz = {0, 0, 0, 0};
#if defined(__clang_major__) && (__clang_major__ >= 23)
        i32x8 gz8 = {0, 0, 0, 0, 0, 0, 0, 0};
        __builtin_amdgcn_tensor_load_to_lds(g0, g1, gz, gz, gz8, 0);
#else
        __builtin_amdgcn_tensor_load_to_lds(g0, g1, gz, gz, 0);
#endif
    }
    __builtin_amdgcn_s_wait_tensorcnt(0);             // s_wait_tensorcnt 0
    __syncthreads();
#else
    // Fallback staging path
    for (int i = threadIdx.x; i < C_ * W_; i += (int)blockDim.x) {
        const int c = i / W_, ww = i % W_;
        s_cur[i] = curRow[(size_t)c * HW_ + ww];
    }
    __syncthreads();
#endif

    // Pull this pixel's 16 current-feature values into registers (DS loads).
    float curReg[C_];
#pragma unroll
    for (int c = 0; c < C_; ++c) curReg[c] = s_cur[c * W_ + w];

    // Per-pixel ray = invK[:3,:3] @ (x, y, 1)
    const float xf = (float)w, yf = (float)h;
    const float* iK = invK + b * 16;
    const float ray0 = iK[0] * xf + iK[1] * yf + iK[2];
    const float ray1 = iK[4] * xf + iK[5] * yf + iK[6];
    const float ray2 = iK[8] * xf + iK[9] * yf + iK[10];

    const float* Kb = Kmat + b * 16;

    // Per frame: P = (K @ T)[:3];  A = P[:,:3] @ ray ; t = P[:,3]
    float Af[F_][3], tf[F_][3];
    bool validf[F_];
#pragma unroll
    for (int f = 0; f < F_; ++f) {
        const float* Tm = poses + ((size_t)(b * F_) + f) * 16;
        float s = 0.f;
#pragma unroll
        for (int k = 0; k < 16; ++k) s += Tm[k];
        validf[f] = (s != 0.0f);
#pragma unroll
        for (int i = 0; i < 3; ++i) {
            const float k0 = Kb[i * 4 + 0], k1 = Kb[i * 4 + 1];
            const float k2 = Kb[i * 4 + 2], k3 = Kb[i * 4 + 3];
            const float p0 = k0 * Tm[0] + k1 * Tm[4] + k2 * Tm[8]  + k3 * Tm[12];
            const float p1 = k0 * Tm[1] + k1 * Tm[5] + k2 * Tm[9]  + k3 * Tm[13];
            const float p2 = k0 * Tm[2] + k1 * Tm[6] + k2 * Tm[10] + k3 * Tm[14];
            const float p3 = k0 * Tm[3] + k1 * Tm[7] + k2 * Tm[11] + k3 * Tm[15];
            Af[f][i] = p0 * ray0 + p1 * ray1 + p2 * ray2;
            tf[f][i] = p3;
        }
    }

    const bool interior = (h >= 2) && (h < H_ - 2) && (w >= 2) && (w < W_ - 2);

    for (int dd = 0; dd < DCHUNK; ++dd) {
        const int d = d0 + dd;
        const float depth = bins[d];
        float sum = 0.f, cnt = 0.f;
        if (interior) {
#pragma unroll
            for (int f = 0; f < F_; ++f) {
                if (!validf[f]) continue;
                const float cx = depth * Af[f][0] + tf[f][0];
                const float cy = depth * Af[f][1] + tf[f][1];
                const float cz = depth * Af[f][2] + tf[f][2] + EPS_;
                const float px = cx / cz;
                const float py = cy / cz;
                const bool edge = (px >= 2.0f) && (px <= (float)(W_ - 2)) &&
                                  (py >= 2.0f) && (py <= (float)(H_ - 2));
                if (!edge) continue;
                // Inside the edge band all four bilinear taps are in-bounds.
                const float x0 = floorf(px), y0 = floorf(py);
                const float fx = px - x0, fy = py - y0;
                const int x0i = (int)x0, y0i = (int)y0;
                const float w00 = (1.f - fx) * (1.f - fy);
                const float w10 = fx * (1.f - fy);
                const float w01 = (1.f - fx) * fy;
                const float w11 = fx * fy;
                const float* lfb = lf + ((size_t)(b * F_ + f)) * C_ * HW_
                                      + (size_t)y0i * W_ + x0i;
                float acc = 0.f;
#pragma unroll
                for (int c = 0; c < C_; ++c) {
                    const float* p = lfb + (size_t)c * HW_;
                    const float t00 = p[0];
                    const float t10 = p[1];
                    const float t01 = p[W_];
                    const float t11 = p[W_ + 1];
                    const float warped = t00 * w00 + t10 * w10 + t01 * w01 + t11 * w11;
                    acc += fabsf(warped - curReg[c]);
                }
                const float diff = acc * (1.0f / (float)C_);
                sum += diff;
                cnt += (diff > 0.0f) ? 1.0f : 0.0f;
            }
        }
        out_cost[(((size_t)b * D_ + d) * H_ + h) * W_ + w] = sum / (cnt + 1e-7f);
    }
}

// ---------------------------------------------------------------------------
// Kernel 2: finalize.  Per (b,h,w): max over D of cost_pre, then
//   missing = (cost==0);  cost = cost*(1-missing) + max*missing
// Reads/writes cost in place (each D-strip owned by exactly one thread),
// writes missing into the second half of d_out.
// ---------------------------------------------------------------------------
__global__ void finalize_kernel(float* __restrict__ out)
{
    const int idx = blockIdx.x * blockDim.x + threadIdx.x;   // over B*H*W
    if (idx >= B_ * HW_) return;
    const int b = idx / HW_;
    const int pix = idx % HW_;

    float* cost = out;
    float* missing = out + (size_t)B_ * D_ * HW_;
    const size_t base = (size_t)b * D_ * HW_ + pix;

    float mx = -3.402823e38f;
#pragma unroll 4
    for (int d = 0; d < D_; ++d)
        mx = fmaxf(mx, cost[base + (size_t)d * HW_]);

#pragma unroll 4
    for (int d = 0; d < D_; ++d) {
        const size_t o = base + (size_t)d * HW_;
        const float cv = cost[o];
        const float miss = (cv == 0.0f) ? 1.0f : 0.0f;
        cost[o] = cv * (1.0f - miss) + mx * miss;
        missing[o] = miss;
    }
}

// ---------------------------------------------------------------------------
extern "C" void kernel_launch(void* const* d_in, const int* in_sizes, int n_in,
                              void* d_out, int out_size, void* d_ws, size_t ws_size,
                              hipStream_t stream)
{
    (void)in_sizes; (void)n_in; (void)out_size; (void)d_ws; (void)ws_size;
    const float* cur   = (const float*)d_in[0];   // [B,C,H,W]
    const float* lf    = (const float*)d_in[1];   // [B,F,C,H,W]
    const float* poses = (const float*)d_in[2];   // [B,F,4,4]
    const float* K     = (const float*)d_in[3];   // [B,4,4]
    const float* invK  = (const float*)d_in[4];   // [B,4,4]
    const float* bins  = (const float*)d_in[5];   // [D]
    float* out = (float*)d_out;                   // cost [B,D,H,W] ++ missing [B,D,H,W]

    dim3 grid1(D_ / DCHUNK, H_, B_);
    dim3 blk1(W_);
    cost_volume_kernel<<<grid1, blk1, 0, stream>>>(cur, lf, poses, K, invK, bins, out);

    const int npix = B_ * H_ * W_;
    finalize_kernel<<<(npix + 255) / 256, 256, 0, stream>>>(out);
}